// CapacityTestMemory_35270271435169
// MI455X (gfx1250) — compile-verified
//
#include <hip/hip_runtime.h>
#include <math.h>

typedef __attribute__((ext_vector_type(2))) float v2f;
typedef __attribute__((ext_vector_type(8))) float v8f;

#define MEM_SLOTS 2048
#define TOPK 8

// ---------------------------------------------------------------------------
// WMMA f32 16x16x4 GEMM, C[M,N] = A[M,K] @ B[K,N] (+bias), B row-major.
// One wave (32 threads) per 16x16 output tile.
// A 16x4 layout: lane L -> row = L&15, holds K = k0 + 2*(L>>4) + {0,1}
// B 4x16 layout: lane L -> col = L&15, holds K = k0 + 2*(L>>4) + {0,1}
// C/D: lane L -> col = L&15, rows = 8*(L>>4) + v, v=0..7
// ---------------------------------------------------------------------------
__global__ __launch_bounds__(32)
void wmma_gemm_rowB(const float* __restrict__ A, const float* __restrict__ B,
                    const float* __restrict__ bias, float* __restrict__ C,
                    int N, int K) {
  const int lane = threadIdx.x;
  const int row  = lane & 15;
  const int col  = lane & 15;
  const int kb   = (lane >> 4) << 1;   // 0 or 2
  const int m0   = blockIdx.y << 4;
  const int n0   = blockIdx.x << 4;

  v8f acc = {};
  const float* ap = A + (size_t)(m0 + row) * K + kb;
  for (int k0 = 0; k0 < K; k0 += 4) {
    v2f a, b;
    a.x = ap[k0];
    a.y = ap[k0 + 1];
    b.x = B[(size_t)(k0 + kb) * N + n0 + col];
    b.y = B[(size_t)(k0 + kb + 1) * N + n0 + col];
    acc = __builtin_amdgcn_wmma_f32_16x16x4_f32(false, a, false, b,
                                                (short)0, acc, false, false);
  }
  const int rbase = (lane >> 4) << 3;
  const float bv = bias ? bias[n0 + col] : 0.0f;
  for (int v = 0; v < 8; ++v) {
    C[(size_t)(m0 + rbase + v) * N + n0 + col] = acc[v] + bv;
  }
}

// Same, but B is stored transposed: Bt is [N,K] row-major; computes A @ Bt^T.
// Used for qk = q @ k_w^T (fold k_w into the query).
__global__ __launch_bounds__(32)
void wmma_gemm_colB(const float* __restrict__ A, const float* __restrict__ Bt,
                    float* __restrict__ C, int N, int K) {
  const int lane = threadIdx.x;
  const int row  = lane & 15;
  const int col  = lane & 15;
  const int kb   = (lane >> 4) << 1;
  const int m0   = blockIdx.y << 4;
  const int n0   = blockIdx.x << 4;

  v8f acc = {};
  const float* ap = A  + (size_t)(m0 + row) * K + kb;
  const float* bp = Bt + (size_t)(n0 + col) * K + kb;   // contiguous K-pair
  for (int k0 = 0; k0 < K; k0 += 4) {
    v2f a, b;
    a.x = ap[k0];     a.y = ap[k0 + 1];
    b.x = bp[k0];     b.y = bp[k0 + 1];
    acc = __builtin_amdgcn_wmma_f32_16x16x4_f32(false, a, false, b,
                                                (short)0, acc, false, false);
  }
  const int rbase = (lane >> 4) << 3;
  for (int v = 0; v < 8; ++v) {
    C[(size_t)(m0 + rbase + v) * N + n0 + col] = acc[v];
  }
}

// ---------------------------------------------------------------------------
// Memory-bound score pass: raw[b,m] = dot(enc[b,pos(m)], qk[b]) / sqrt(H)
// (the uniform q·k_b shift is dropped: it cannot change top-k nor softmax).
// One wave per slot, 8 slots per 256-thread block; qk[b] staged in LDS.
// ---------------------------------------------------------------------------
__global__ __launch_bounds__(256)
void score_kernel(const float* __restrict__ enc, const float* __restrict__ qk,
                  const int* __restrict__ np_p, float* __restrict__ raw,
                  int T, int H) {
  __shared__ __align__(16) float s_qk[512];
  const int b = blockIdx.y;
  for (int i = threadIdx.x; i < H; i += 256) s_qk[i] = qk[(size_t)b * H + i];
  __syncthreads();

  const int wave = threadIdx.x >> 5;
  const int lane = threadIdx.x & 31;
  const int m = (blockIdx.x << 3) + wave;
  const int np = *np_p;
  const int L = min(2 * np, T - 3);

  float sum = 0.0f;
  if (m < L) {
    int pos = m + MEM_SLOTS * ((L - 1 - m) / MEM_SLOTS);
    pos = min(max(pos, 0), T - 1);
    const float4* rowp = (const float4*)(enc + ((size_t)b * T + pos) * H);
    const float4* qkp  = (const float4*)s_qk;
    const int nv = H >> 2;                 // float4 count (128)
    for (int i = lane; i < nv; i += 32) {  // coalesced b128 streams
      float4 e = rowp[i];
      float4 w = qkp[i];
      sum += e.x * w.x + e.y * w.y + e.z * w.z + e.w * w.w;
    }
    for (int off = 16; off > 0; off >>= 1) sum += __shfl_xor(sum, off, 32);
  }
  if (lane == 0)
    raw[(size_t)b * MEM_SLOTS + m] = (m < L) ? sum * rsqrtf((float)H) : 0.0f;
}

// ---------------------------------------------------------------------------
// Top-8 + softmax per batch row of 2048 scores (LDS argmax, 8 rounds).
// ---------------------------------------------------------------------------
__global__ __launch_bounds__(256)
void topk_kernel(const float* __restrict__ raw, float* __restrict__ probs,
                 int* __restrict__ idxs) {
  __shared__ float s_val[MEM_SLOTS];
  __shared__ float r_val[256];
  __shared__ int   r_idx[256];
  __shared__ float sel_val[TOPK];
  __shared__ int   sel_idx[TOPK];
  const int b = blockIdx.x;
  const int t = threadIdx.x;
  for (int i = t; i < MEM_SLOTS; i += 256) s_val[i] = raw[(size_t)b * MEM_SLOTS + i];
  __syncthreads();

  for (int k = 0; k < TOPK; ++k) {
    float best = -3.0e38f; int bi = 0x7fffffff;
    for (int i = t; i < MEM_SLOTS; i += 256) {
      float v = s_val[i];
      if (v > best || (v == best && i < bi)) { best = v; bi = i; }
    }
    r_val[t] = best; r_idx[t] = bi;
    __syncthreads();
    for (int s = 128; s > 0; s >>= 1) {
      if (t < s) {
        float ov = r_val[t + s]; int oi = r_idx[t + s];
        if (ov > r_val[t] || (ov == r_val[t] && oi < r_idx[t])) {
          r_val[t] = ov; r_idx[t] = oi;
        }
      }
      __syncthreads();
    }
    if (t == 0) { sel_val[k] = r_val[0]; sel_idx[k] = r_idx[0]; s_val[r_idx[0]] = -3.0e38f; }
    __syncthreads();
  }
  if (t == 0) {
    float mx = sel_val[0];      // selections are in descending order
    float e[TOPK], sum = 0.0f;
    for (int k = 0; k < TOPK; ++k) { e[k] = expf(sel_val[k] - mx); sum += e[k]; }
    float inv = 1.0f / sum;
    for (int k = 0; k < TOPK; ++k) {
      probs[b * TOPK + k] = e[k] * inv;
      idxs [b * TOPK + k] = sel_idx[k];
    }
  }
}

// ---------------------------------------------------------------------------
// retrieved + query: gather the 8 selected memory rows (invalid slots are
// zero rows in the reference -> skipped), weighted sum, add query.
// ---------------------------------------------------------------------------
__global__ __launch_bounds__(128)
void gather_kernel(const float* __restrict__ enc, const float* __restrict__ query,
                   const float* __restrict__ probs, const int* __restrict__ idxs,
                   const int* __restrict__ np_p, float* __restrict__ out_in,
                   int T, int H) {
  const int b = blockIdx.x;
  const int t = threadIdx.x;           // H/4 threads, one float4 each
  const int np = *np_p;
  const int L = min(2 * np, T - 3);
  float4 acc = make_float4(0.f, 0.f, 0.f, 0.f);
  for (int k = 0; k < TOPK; ++k) {
    int m = idxs[b * TOPK + k];
    if (m >= L) continue;              // masked (zero) memory row
    float p = probs[b * TOPK + k];
    int pos = m + MEM_SLOTS * ((L - 1 - m) / MEM_SLOTS);
    pos = min(max(pos, 0), T - 1);
    float4 e = ((const float4*)(enc + ((size_t)b * T + pos) * H))[t];
    acc.x += p * e.x; acc.y += p * e.y; acc.z += p * e.z; acc.w += p * e.w;
  }
  float4 q = ((const float4*)(query + (size_t)b * H))[t];
  acc.x += q.x; acc.y += q.y; acc.z += q.z; acc.w += q.w;
  ((float4*)(out_in + (size_t)b * H))[t] = acc;
}

// ---------------------------------------------------------------------------
extern "C" void kernel_launch(void* const* d_in, const int* in_sizes, int n_in,
                              void* d_out, int out_size, void* d_ws, size_t ws_size,
                              hipStream_t stream) {
  const float* enc   = (const float*)d_in[0];
  const float* query = (const float*)d_in[1];
  const int*   np_p  = (const int*)  d_in[2];
  const float* q_w   = (const float*)d_in[3];
  const float* q_b   = (const float*)d_in[4];
  const float* k_w   = (const float*)d_in[5];
  /* k_b (d_in[6]) provably cannot affect the output: it adds a per-batch
     constant to every score, invariant under top-k + softmax, and the
     reference's k_b never touches `memory` or `retrieved`. */
  const float* out_w = (const float*)d_in[7];
  const float* out_b = (const float*)d_in[8];

  const int H = in_sizes[4];                 // 512
  const int B = in_sizes[1] / H;             // 32
  const int T = in_sizes[0] / (B * H);       // 4096
  const int V = in_sizes[8];                 // 128

  float* ws     = (float*)d_ws;
  float* q      = ws;                         // B*H
  float* qk     = q   + (size_t)B * H;        // B*H
  float* raw    = qk  + (size_t)B * H;        // B*MEM_SLOTS
  float* probs  = raw + (size_t)B * MEM_SLOTS;// B*TOPK
  int*   idxs   = (int*)(probs + B * TOPK);   // B*TOPK
  float* out_in = (float*)(idxs + B * TOPK);  // B*H
  (void)ws_size; (void)n_in; (void)out_size;

  // 1) q = query @ q_w + q_b            (WMMA f32, 32x512x512)
  wmma_gemm_rowB<<<dim3(H / 16, B / 16), 32, 0, stream>>>(query, q_w, q_b, q, H, H);

  // 2) qk = q @ k_w^T                   (WMMA f32, fold k_w into query)
  wmma_gemm_colB<<<dim3(H / 16, B / 16), 32, 0, stream>>>(q, k_w, qk, H, H);

  // 3) raw scores: one streaming pass over the gathered rows (HBM-bound)
  score_kernel<<<dim3(MEM_SLOTS / 8, B), 256, 0, stream>>>(enc, qk, np_p, raw, T, H);

  // 4) top-8 + softmax per batch
  topk_kernel<<<dim3(B), 256, 0, stream>>>(raw, probs, idxs);

  // 5) retrieved + query
  gather_kernel<<<dim3(B), H / 4, 0, stream>>>(enc, query, probs, idxs, np_p, out_in, T, H);

  // 6) logits = out_in @ out_w + out_b  (WMMA f32, 32x512x128)
  wmma_gemm_rowB<<<dim3(V / 16, B / 16), 32, 0, stream>>>(out_in, out_w, out_b,
                                                          (float*)d_out, V, H);
}